// residualBlock_57148834840959
// MI455X (gfx1250) — compile-verified
//
#include <hip/hip_runtime.h>
#include <stdint.h>

typedef float v2f __attribute__((ext_vector_type(2)));
typedef float v8f __attribute__((ext_vector_type(8)));

#define N_VOX  262144
#define HCH    64
#define K_OFF  27
#define M_TILE 128
#define LDA    68   // LDS row stride (floats) for gathered A tile: conflict-free A frag b64 reads
#define LDB    72   // LDS row stride (floats) for W tile: 2*LDB % 64 == 16 -> halves on disjoint banks

// raw LDS byte offset of a __shared__ element (flat shared aperture keeps offset in addr[31:0])
__device__ __forceinline__ uint32_t lds_addr(const void* p) {
    return (uint32_t)(uintptr_t)p;
}

// Fused gather -> GEMM(27 offsets) -> epilogue, double-buffered via async global->LDS DMA.
//   out[r,:] = epilogue( sum_k  in[nbr[r,k], :] @ W[k] )
__global__ __launch_bounds__(256)
void spconv_wmma_f32(const float* __restrict__ in,    // [N,64] gather source
                     const float* __restrict__ W,     // [27,64,64]
                     const int*   __restrict__ nbr,   // [N,27]
                     const float* __restrict__ resid, // nullptr or [N,64]
                     float* __restrict__ out,         // [N,64]
                     int relu)
{
    __shared__ float lds_a[2][M_TILE * LDA];  // 2 x 34,816 B
    __shared__ float lds_w[2][HCH * LDB];     // 2 x 18,432 B

    const int tid  = threadIdx.x;
    const int wave = tid >> 5;     // 0..7 : 16-row block
    const int lane = tid & 31;
    const int h    = lane >> 4;
    const int ml   = lane & 15;
    const int rowBase = blockIdx.x * M_TILE;

    // gather assignment: 2 threads per row, each owns a contiguous 128B half-row
    const int grow  = tid >> 1;          // 0..127
    const int ghalf = (tid & 1) * 32;    // float offset within 64-float row
    const long gRowIdx = (long)(rowBase + grow) * K_OFF;

    // ---- async issue helpers (each wave issues exactly 12 async ops per batch) ----
    auto issue_gather = [&](int nb, int idx) {
        const long idx_safe = (idx < N_VOX) ? idx : 0;   // sentinel -> safe addr, zero-fixed later
        const uint64_t ga = (uint64_t)(uintptr_t)(in + idx_safe * HCH + ghalf);
        const uint32_t la = lds_addr(&lds_a[nb][grow * LDA + ghalf]);
        asm volatile(
            "global_load_async_to_lds_b128 %0, %1, off\n\t"
            "global_load_async_to_lds_b128 %0, %1, off offset:16\n\t"
            "global_load_async_to_lds_b128 %0, %1, off offset:32\n\t"
            "global_load_async_to_lds_b128 %0, %1, off offset:48\n\t"
            "global_load_async_to_lds_b128 %0, %1, off offset:64\n\t"
            "global_load_async_to_lds_b128 %0, %1, off offset:80\n\t"
            "global_load_async_to_lds_b128 %0, %1, off offset:96\n\t"
            "global_load_async_to_lds_b128 %0, %1, off offset:112"
            :: "v"(la), "v"(ga) : "memory");
    };
    auto issue_weights = [&](int nb, int k) {
        const float* wsrcK = W + k * HCH * HCH;
        #pragma unroll
        for (int j = 0; j < 4; ++j) {
            const int g  = tid + j * 256;        // float4 index 0..1023
            const int r  = g >> 4;
            const int c4 = (g & 15) * 4;
            const uint64_t ga = (uint64_t)(uintptr_t)(wsrcK + g * 4);
            const uint32_t la = lds_addr(&lds_w[nb][r * LDB + c4]);
            asm volatile("global_load_async_to_lds_b128 %0, %1, off"
                         :: "v"(la), "v"(ga) : "memory");
        }
    };

    v8f acc0 = {}, acc1 = {}, acc2 = {}, acc3 = {};

    // ---- software pipeline prologue: batch 0 in flight, idx stream 2 ahead ----
    int idx_cur  = nbr[gRowIdx + 0];
    issue_gather(0, idx_cur);
    issue_weights(0, 0);
    int idx_next = nbr[gRowIdx + 1];

    for (int k = 0; k < K_OFF; ++k) {
        const int cur = k & 1;

        const int idx_fix = idx_cur;             // sentinel state of the batch we consume now
        if (k + 1 < K_OFF) {
            // other buffer was last read at compute(k-1); barrier at loop end protects it
            issue_gather(1 - cur, idx_next);
            issue_weights(1 - cur, k + 1);
            idx_cur = idx_next;
            if (k + 2 < K_OFF) idx_next = nbr[gRowIdx + k + 2];  // latency hides under compute
            asm volatile("s_wait_asynccnt 12" ::: "memory");     // drain batch k (in-order retire)
        } else {
            asm volatile("s_wait_asynccnt 0" ::: "memory");
        }

        // sentinel row fix-up: overwrite my half-row with zeros (after async data landed)
        if (idx_fix >= N_VOX) {
            const float4 z = {0.f, 0.f, 0.f, 0.f};
            float4* dz = (float4*)(&lds_a[cur][grow * LDA + ghalf]);
            #pragma unroll
            for (int j = 0; j < 8; ++j) dz[j] = z;
        }
        __syncthreads();   // all waves' batch-k data (and zero fixes) visible

        // ---- 16x64 @ 64x64 tile GEMM via V_WMMA_F32_16X16X4_F32 ----
        const float* aRow = &lds_a[cur][(wave * 16 + ml) * LDA];
        const float* wBuf = &lds_w[cur][0];
        #pragma unroll
        for (int kk = 0; kk < 16; ++kk) {
            const int kc0 = kk * 4;
            const v2f a = *(const v2f*)(aRow + kc0 + 2 * h);       // A: VGPR0=K(kc0+2h), VGPR1=+1
            const int brow = (kc0 + 2 * h) * LDB + ml;             // B: row K=kc0+2h, col N=ml
            {
                v2f b; b.x = wBuf[brow +  0]; b.y = wBuf[brow + LDB +  0];
                acc0 = __builtin_amdgcn_wmma_f32_16x16x4_f32(false, a, false, b, (short)0, acc0, false, false);
            }
            {
                v2f b; b.x = wBuf[brow + 16]; b.y = wBuf[brow + LDB + 16];
                acc1 = __builtin_amdgcn_wmma_f32_16x16x4_f32(false, a, false, b, (short)0, acc1, false, false);
            }
            {
                v2f b; b.x = wBuf[brow + 32]; b.y = wBuf[brow + LDB + 32];
                acc2 = __builtin_amdgcn_wmma_f32_16x16x4_f32(false, a, false, b, (short)0, acc2, false, false);
            }
            {
                v2f b; b.x = wBuf[brow + 48]; b.y = wBuf[brow + LDB + 48];
                acc3 = __builtin_amdgcn_wmma_f32_16x16x4_f32(false, a, false, b, (short)0, acc3, false, false);
            }
        }
        __syncthreads();   // compute(k) done before batch k+2 overwrites buf[cur]
    }

    // ---- epilogue: C/D layout VGPR j -> M = j + 8*h, N = ml ----
    v8f accs[4] = {acc0, acc1, acc2, acc3};
    #pragma unroll
    for (int cb = 0; cb < 4; ++cb) {
        #pragma unroll
        for (int j = 0; j < 8; ++j) {
            const long r = rowBase + wave * 16 + j + 8 * h;
            const long c = cb * 16 + ml;
            float v = accs[cb][j];
            if (relu) v = v > 0.f ? v : 0.f;
            if (resid) v += resid[r * HCH + c];
            out[r * HCH + c] = v;
        }
    }
}

extern "C" void kernel_launch(void* const* d_in, const int* in_sizes, int n_in,
                              void* d_out, int out_size, void* d_ws, size_t ws_size,
                              hipStream_t stream) {
    const float* x   = (const float*)d_in[0];   // [N,64]
    const float* Wa  = (const float*)d_in[1];   // [27,64,64]
    const float* Wb  = (const float*)d_in[2];   // [27,64,64]
    const int*   nbr = (const int*)d_in[3];     // [N,27]
    float* mid = (float*)d_ws;                  // [N,64] intermediate (64 MB)
    float* out = (float*)d_out;

    dim3 grid(N_VOX / M_TILE), block(256);
    // pass 1: relu(spconv(x, Wa))
    spconv_wmma_f32<<<grid, block, 0, stream>>>(x, Wa, nbr, nullptr, mid, 1);
    // pass 2: spconv(mid, Wb) + x
    spconv_wmma_f32<<<grid, block, 0, stream>>>(mid, Wb, nbr, x, out, 0);
}